// AttentionHead_17377437680201
// MI455X (gfx1250) — compile-verified
//
#include <hip/hip_runtime.h>
#include <math.h>

typedef float v2f __attribute__((ext_vector_type(2)));
typedef float v8f __attribute__((ext_vector_type(8)));

#define B_  8
#define S_  2048
#define D_  512
#define DK_ 64

// D = A(16x4) * B(4x16) + C, fp32 WMMA (CDNA5 / gfx1250)
__device__ __forceinline__ v8f wmma4(v2f a, v2f b, v8f c) {
    return __builtin_amdgcn_wmma_f32_16x16x4_f32(
        /*neg_a=*/false, a, /*neg_b=*/false, b,
        /*c_mod=*/(short)0, c, /*reuse_a=*/false, /*reuse_b=*/false);
}

// ---------------------------------------------------------------------------
// Kernel 1: fused projections  out = X @ W + b   (X: [16384,512], W: [512,64])
// One wave computes a 16x64 output tile. blockIdx.y selects q/k/v.
// HBM-bound (101 MB of activations); WMMA overlapped with load clauses.
// ---------------------------------------------------------------------------
__global__ __launch_bounds__(256)
void proj_kernel(const float* __restrict__ q, const float* __restrict__ k,
                 const float* __restrict__ v,
                 const float* __restrict__ Wq, const float* __restrict__ bq,
                 const float* __restrict__ Wk, const float* __restrict__ bk,
                 const float* __restrict__ Wv, const float* __restrict__ bv,
                 float* __restrict__ qp, float* __restrict__ kp,
                 float* __restrict__ vp)
{
    const int lane = threadIdx.x & 31;
    const int wave = threadIdx.x >> 5;
    const int n  = lane & 15;   // column-within-tile / row-within-A-tile
    const int hi = lane >> 4;   // half-wave select

    const float* A; const float* W; const float* bias; float* Out;
    if (blockIdx.y == 0)      { A = q; W = Wq; bias = bq; Out = qp; }
    else if (blockIdx.y == 1) { A = k; W = Wk; bias = bk; Out = kp; }
    else                      { A = v; W = Wv; bias = bv; Out = vp; }

    const int m0 = (blockIdx.x * 8 + wave) * 16;   // 16 rows per wave
    const float* __restrict__ arow = A + (size_t)(m0 + n) * D_;

    v8f acc0 = {}, acc1 = {}, acc2 = {}, acc3 = {};

    for (int k0 = 0; k0 < D_; k0 += 4) {
        const int kk = k0 + 2 * hi;
        v2f a; a.x = arow[kk]; a.y = arow[kk + 1];

        const float* __restrict__ w0 = W + (size_t)kk * DK_;
        const float* __restrict__ w1 = w0 + DK_;
        v2f b0; b0.x = w0[n];      b0.y = w1[n];
        v2f b1; b1.x = w0[16 + n]; b1.y = w1[16 + n];
        v2f b2; b2.x = w0[32 + n]; b2.y = w1[32 + n];
        v2f b3; b3.x = w0[48 + n]; b3.y = w1[48 + n];

        acc0 = wmma4(a, b0, acc0);
        acc1 = wmma4(a, b1, acc1);
        acc2 = wmma4(a, b2, acc2);
        acc3 = wmma4(a, b3, acc3);
    }

    const float bb0 = bias[n], bb1 = bias[16 + n], bb2 = bias[32 + n], bb3 = bias[48 + n];
#pragma unroll
    for (int j = 0; j < 8; ++j) {
        const size_t row = (size_t)(m0 + j + 8 * hi) * DK_;
        Out[row + n]      = acc0[j] + bb0;
        Out[row + 16 + n] = acc1[j] + bb1;
        Out[row + 32 + n] = acc2[j] + bb2;
        Out[row + 48 + n] = acc3[j] + bb3;
    }
}

// ---------------------------------------------------------------------------
// Kernel 2: causal flash attention, fp32 WMMA, split-K over 4 waves.
// One workgroup (4 waves) per 16-row query tile; wave w handles key tiles
// kt = w, w+4, ... <= qt with private online-softmax state, then the four
// partials are merged in LDS with a log-sum-exp combine (exact, fp32).
// ---------------------------------------------------------------------------
__global__ __launch_bounds__(128)
void attn_kernel(const float* __restrict__ qp, const float* __restrict__ kp,
                 const float* __restrict__ vp, const float* __restrict__ pad,
                 float* __restrict__ out)
{
    const int lane = threadIdx.x & 31;
    const int wave = threadIdx.x >> 5;
    const int n  = lane & 15;
    const int hi = lane >> 4;

    const int t  = blockIdx.x;                 // 0..1023 query tiles
    const int b  = t >> 7;                     // batch
    const int qt = t & 127;                    // query tile within batch
    const int q0 = qt * 16;

    const float* __restrict__ qb = qp + (size_t)b * S_ * DK_;
    const float* __restrict__ kb = kp + (size_t)b * S_ * DK_;
    const float* __restrict__ vb = vp + (size_t)b * S_ * DK_;
    const float* __restrict__ pm = pad + (size_t)b * S_;

    __shared__ float pbuf[4][16 * 17];         // P staging (C->A relayout), padded
    __shared__ float o_lds[4][16 * 68];        // per-wave partial O, padded stride
    __shared__ float ml_lds[4][16][2];         // per-wave partial (m, l) per row
    float* pl = pbuf[wave];

    // Preload Q tile in WMMA A-layout: 16 k-steps of float2 per lane.
    v2f qa[16];
    {
        const float* __restrict__ qrow = qb + (size_t)(q0 + n) * DK_;
#pragma unroll
        for (int s4 = 0; s4 < 16; ++s4) {
            const int kk = s4 * 4 + 2 * hi;
            qa[s4].x = qrow[kk]; qa[s4].y = qrow[kk + 1];
        }
    }

    v8f o0 = {}, o1 = {}, o2 = {}, o3 = {};
    float m_r[8], l_r[8];
#pragma unroll
    for (int j = 0; j < 8; ++j) { m_r[j] = -1e30f; l_r[j] = 0.0f; }

    const float scale = 0.125f;   // 1/sqrt(64)

    for (int kt = wave; kt <= qt; kt += 4) {
        const int kbase = kt * 16;

        // Prefetch this wave's next key/value tiles (global_prefetch_b8).
        if (kt + 4 <= qt) {
            __builtin_prefetch(kb + (size_t)(kbase + 64 + n) * DK_ + 4 * hi, 0, 1);
            __builtin_prefetch(vb + (size_t)(kbase + 64 + n) * DK_ + 4 * hi, 0, 1);
        }

        // ---- scores S = Q @ K^T (16x16, fp32 accum) ----
        v8f s = {};
        {
            const float* __restrict__ krow = kb + (size_t)(kbase + n) * DK_;
#pragma unroll
            for (int s4 = 0; s4 < 16; ++s4) {
                const int kk = s4 * 4 + 2 * hi;
                v2f bbv; bbv.x = krow[kk]; bbv.y = krow[kk + 1];
                s = wmma4(qa[s4], bbv, s);
            }
        }

        const float pmv = pm[kbase + n];
        const int col = kbase + n;

        // ---- scale + causal mask + pad mask ----
        float p[8], mnew[8], alpha[8];
#pragma unroll
        for (int j = 0; j < 8; ++j) {
            const int row = q0 + j + 8 * hi;
            p[j] = (col > row) ? -1e30f : (s[j] * scale + pmv);
        }

        // ---- row max across the 16 lanes holding this row ----
#pragma unroll
        for (int j = 0; j < 8; ++j) {
            float mx = p[j];
            mx = fmaxf(mx, __shfl_xor(mx, 1, 32));
            mx = fmaxf(mx, __shfl_xor(mx, 2, 32));
            mx = fmaxf(mx, __shfl_xor(mx, 4, 32));
            mx = fmaxf(mx, __shfl_xor(mx, 8, 32));
            mnew[j] = fmaxf(m_r[j], mx);
        }

        // ---- exp, row sum, online rescale ----
#pragma unroll
        for (int j = 0; j < 8; ++j) {
            const float e = __expf(p[j] - mnew[j]);
            p[j] = e;
            alpha[j] = __expf(m_r[j] - mnew[j]);
            m_r[j] = mnew[j];
            float rs = e;
            rs += __shfl_xor(rs, 1, 32);
            rs += __shfl_xor(rs, 2, 32);
            rs += __shfl_xor(rs, 4, 32);
            rs += __shfl_xor(rs, 8, 32);
            l_r[j] = l_r[j] * alpha[j] + rs;
        }

        // ---- relayout P (C-format -> A-format) through LDS ----
#pragma unroll
        for (int j = 0; j < 8; ++j)
            pl[(j + 8 * hi) * 17 + n] = p[j];
        asm volatile("s_wait_dscnt 0" ::: "memory");

#pragma unroll
        for (int j = 0; j < 8; ++j) {
            o0[j] *= alpha[j]; o1[j] *= alpha[j];
            o2[j] *= alpha[j]; o3[j] *= alpha[j];
        }

        // ---- O += P @ V (16x16 @ 16x64) ----
#pragma unroll
        for (int s4 = 0; s4 < 4; ++s4) {
            const int kk = s4 * 4 + 2 * hi;       // local key index
            v2f pa; pa.x = pl[n * 17 + kk]; pa.y = pl[n * 17 + kk + 1];
            const float* __restrict__ v0 = vb + (size_t)(kbase + kk) * DK_;
            const float* __restrict__ v1 = v0 + DK_;
            v2f b0; b0.x = v0[n];      b0.y = v1[n];
            v2f b1; b1.x = v0[16 + n]; b1.y = v1[16 + n];
            v2f b2; b2.x = v0[32 + n]; b2.y = v1[32 + n];
            v2f b3; b3.x = v0[48 + n]; b3.y = v1[48 + n];
            o0 = wmma4(pa, b0, o0);
            o1 = wmma4(pa, b1, o1);
            o2 = wmma4(pa, b2, o2);
            o3 = wmma4(pa, b3, o3);
        }
    }

    // ---- publish this wave's partial (O', m', l') to LDS ----
    {
        float* __restrict__ ow = o_lds[wave];
#pragma unroll
        for (int j = 0; j < 8; ++j) {
            const int r = j + 8 * hi;
            ow[r * 68 + n]      = o0[j];
            ow[r * 68 + 16 + n] = o1[j];
            ow[r * 68 + 32 + n] = o2[j];
            ow[r * 68 + 48 + n] = o3[j];
            if (n == 0) {
                ml_lds[wave][r][0] = m_r[j];
                ml_lds[wave][r][1] = l_r[j];
            }
        }
    }
    __syncthreads();

    // ---- merge the 4 split-K partials: O = sum_w O'_w * e^(m'_w - m) ----
    // 128 threads: thread -> (row = tid/8, 8 columns).
    {
        const int tid = threadIdx.x;
        const int row = tid >> 3;
        const int c0  = (tid & 7) * 8;

        const float m0w = ml_lds[0][row][0];
        const float m1w = ml_lds[1][row][0];
        const float m2w = ml_lds[2][row][0];
        const float m3w = ml_lds[3][row][0];
        const float gm = fmaxf(fmaxf(m0w, m1w), fmaxf(m2w, m3w));
        const float e0 = __expf(m0w - gm);
        const float e1 = __expf(m1w - gm);
        const float e2 = __expf(m2w - gm);
        const float e3 = __expf(m3w - gm);
        const float lsum = ml_lds[0][row][1] * e0 + ml_lds[1][row][1] * e1 +
                           ml_lds[2][row][1] * e2 + ml_lds[3][row][1] * e3;
        const float inv = 1.0f / (lsum + 1e-10f);   // matches reference denom

        float* __restrict__ ob = out + (size_t)b * S_ * DK_ + (size_t)(q0 + row) * DK_;
        const float* __restrict__ p0 = &o_lds[0][row * 68 + c0];
        const float* __restrict__ p1 = &o_lds[1][row * 68 + c0];
        const float* __restrict__ p2 = &o_lds[2][row * 68 + c0];
        const float* __restrict__ p3 = &o_lds[3][row * 68 + c0];
#pragma unroll
        for (int c = 0; c < 8; ++c) {
            const float acc = p0[c] * e0 + p1[c] * e1 + p2[c] * e2 + p3[c] * e3;
            ob[c0 + c] = acc * inv;
        }
    }
}

// ---------------------------------------------------------------------------
extern "C" void kernel_launch(void* const* d_in, const int* in_sizes, int n_in,
                              void* d_out, int out_size, void* d_ws, size_t ws_size,
                              hipStream_t stream) {
    const float* q   = (const float*)d_in[0];
    const float* k   = (const float*)d_in[1];
    const float* v   = (const float*)d_in[2];
    const float* pad = (const float*)d_in[3];
    const float* Wq  = (const float*)d_in[4];
    const float* bq  = (const float*)d_in[5];
    const float* Wk  = (const float*)d_in[6];
    const float* bk  = (const float*)d_in[7];
    const float* Wv  = (const float*)d_in[8];
    const float* bv  = (const float*)d_in[9];
    float* out = (float*)d_out;

    float* qp = (float*)d_ws;                    // [8,2048,64]
    float* kp = qp + (size_t)B_ * S_ * DK_;
    float* vp = kp + (size_t)B_ * S_ * DK_;      // 12 MB total workspace

    // Projections: 1024 row-tiles per matrix, 8 waves (tiles) per 256-thread block.
    dim3 g1(128, 3), b1(256);
    proj_kernel<<<g1, b1, 0, stream>>>(q, k, v, Wq, bq, Wk, bk, Wv, bv, qp, kp, vp);

    // Flash attention: one 4-wave workgroup per query tile, split-K over waves.
    attn_kernel<<<dim3(1024), dim3(128), 0, stream>>>(qp, kp, vp, pad, out);
}